// PositionWeightedModuleCollection_81423989997924
// MI455X (gfx1250) — compile-verified
//
#include <hip/hip_runtime.h>
#include <stdint.h>

// ---------------------------------------------------------------------------
// PositionWeightedModuleCollection gather:
//   out[t] = position_weights[key_ids[t] * L + positions[t]],  T = 25M, L = 200
//
// Pure bandwidth problem: 12 B/elem (two i32 reads + one f32 write) = 300 MB
// => ~13 us floor at 23.3 TB/s. Table (6.4 KB) lives in LDS; index/output
// streams use 128-bit non-temporal VMEM. Table is staged via the CDNA5
// Tensor Data Mover (tensor_load_to_lds + s_wait_tensorcnt) when available.
// ---------------------------------------------------------------------------

typedef int          __attribute__((ext_vector_type(4))) i32x4;
typedef float        __attribute__((ext_vector_type(4))) f32x4;
typedef unsigned int __attribute__((ext_vector_type(4))) u32x4;
typedef int          __attribute__((ext_vector_type(8))) i32x8;

#define PW_L    200      // positions table length per key (fixed by module)
#define TAB_MAX 2048     // LDS table capacity (>= F*L = 1600), 8 KB
#define BLOCK   256      // 8 waves per workgroup on wave32

#if defined(__has_builtin)
#  if __has_builtin(__builtin_amdgcn_tensor_load_to_lds) && \
      __has_builtin(__builtin_amdgcn_s_wait_tensorcnt)
#    define HAVE_TDM 1
#  endif
#endif
#ifndef HAVE_TDM
#  define HAVE_TDM 0
#endif

__global__ __launch_bounds__(BLOCK) void pw_gather_kernel(
    const float* __restrict__ pw,
    const int*   __restrict__ keys,
    const int*   __restrict__ pos,
    float*       __restrict__ out,
    int tabN, int n)
{
    __shared__ float tab[TAB_MAX];

    int nelem = tabN < TAB_MAX ? tabN : TAB_MAX;

#if HAVE_TDM
    // Stage the (tiny) weight table into LDS with the Tensor Data Mover.
    // 1-D tensor: tile_dim0 = tensor_dim0 = nelem f32 elements.
    if (threadIdx.x == 0) {
        uint64_t ga  = (uint64_t)(uintptr_t)pw;
        uint32_t lds = (uint32_t)(uintptr_t)(void*)tab; // low 32b of generic ptr = LDS byte offset
        uint32_t ne  = (uint32_t)nelem;

        u32x4 g0;
        g0.x = 1u;                                   // count=1 (valid), is_restore=0, gather off
        g0.y = lds;                                  // lds_addr (bytes)
        g0.z = (uint32_t)ga;                         // global_addr[31:0]
        g0.w = ((uint32_t)(ga >> 32) & 0x01FFFFFFu)  // global_addr[56:32]
             | (2u << 30);                           // type = 2 ("image")

        i32x8 g1;
        g1[0] = (int)(2u << 16);                     // workgroup_mask=0, data_size=2 (4 bytes)
        g1[1] = (int)((ne & 0xFFFFu) << 16);         // tensor_dim0[15:0] in bits [63:48]
        g1[2] = (int)(((ne >> 16) & 0xFFFFu)         // tensor_dim0[31:16]
             |  (1u << 16));                         // tensor_dim1[15:0] = 1
        g1[3] = (int)((ne & 0xFFFFu) << 16);         // tensor_dim1[31:16]=0, tile_dim0 = ne
        g1[4] = 1;                                   // tile_dim1 = 1, tile_dim2 = 0
        g1[5] = (int)ne;                             // tensor_dim0_stride[31:0]
        g1[6] = 0;                                   // stride hi / tensor_dim1_stride lo
        g1[7] = 0;

        i32x4 z4 = {0, 0, 0, 0};                     // groups 2/3 unused (<=2D tensor)
        i32x8 z8 = {0, 0, 0, 0, 0, 0, 0, 0};         // extra group (6-arg toolchain form)
        __builtin_amdgcn_tensor_load_to_lds(g0, g1, z4, z4, z8, 0);
        __builtin_amdgcn_s_wait_tensorcnt(0);
    }
#else
    for (int i = threadIdx.x; i < nelem; i += BLOCK)
        tab[i] = pw[i];
#endif
    __syncthreads();

    const int tid    = blockIdx.x * BLOCK + threadIdx.x;
    const int stride = gridDim.x * BLOCK;
    const int n4     = n >> 2;

    const i32x4* __restrict__ k4 = (const i32x4*)keys;
    const i32x4* __restrict__ p4 = (const i32x4*)pos;
    f32x4*       __restrict__ o4 = (f32x4*)out;

    // Main stream: 128-bit non-temporal loads/stores, LDS gather in between.
    for (int i = tid; i < n4; i += stride) {
        i32x4 k = __builtin_nontemporal_load(&k4[i]);
        i32x4 p = __builtin_nontemporal_load(&p4[i]);
        f32x4 r;
        r.x = tab[k.x * PW_L + p.x];
        r.y = tab[k.y * PW_L + p.y];
        r.z = tab[k.z * PW_L + p.z];
        r.w = tab[k.w * PW_L + p.w];
        __builtin_nontemporal_store(r, &o4[i]);
    }

    // Scalar tail (n not divisible by 4).
    for (int i = (n4 << 2) + tid; i < n; i += stride)
        out[i] = tab[keys[i] * PW_L + pos[i]];
}

extern "C" void kernel_launch(void* const* d_in, const int* in_sizes, int n_in,
                              void* d_out, int out_size, void* d_ws, size_t ws_size,
                              hipStream_t stream) {
    (void)n_in; (void)out_size; (void)d_ws; (void)ws_size;

    const float* pw   = (const float*)d_in[0];  // position_weights, F*L f32
    const int*   keys = (const int*)d_in[1];    // key_ids, T i32
    const int*   pos  = (const int*)d_in[2];    // positions, T i32
    float*       out  = (float*)d_out;          // T f32

    const int tabN = in_sizes[0];
    const int n    = in_sizes[1];

    int n4     = n >> 2;
    int blocks = (n4 + BLOCK - 1) / BLOCK;
    if (blocks > 8192) blocks = 8192;   // grid-stride: amortize table staging
    if (blocks < 1)    blocks = 1;

    pw_gather_kernel<<<blocks, BLOCK, 0, stream>>>(pw, keys, pos, out, tabN, n);
}